// HcPost_41798621724794
// MI455X (gfx1250) — compile-verified
//
#include <hip/hip_runtime.h>
#include <stdint.h>

// Problem constants (from the reference): B,S,M,N,D
#define Bc 4
#define Sc 4096
#define Mc 4
#define Nc 4
#define Dc 2048

typedef unsigned int u32x4 __attribute__((ext_vector_type(4)));
typedef int          i32x8 __attribute__((ext_vector_type(8)));
typedef int          i32x4 __attribute__((ext_vector_type(4)));
typedef float        f32x4 __attribute__((ext_vector_type(4)));

// ---------------------------------------------------------------------------
// Tensor Data Mover: async DMA of a 2D f32 tile (dim0 elems x dim1 rows,
// row stride stride0 elems) from global memory into LDS at lds_byte_addr.
// D# layout per CDNA5 ISA ch.8 (groups 0..3). Not in a cluster =>
// workgroup_mask must be 0. Tracked with TENSORcnt.
// ---------------------------------------------------------------------------
__device__ __forceinline__ void tdm_load_2d_f32(uint32_t lds_byte_addr,
                                                const float* gptr,
                                                uint32_t dim0, uint32_t dim1,
                                                uint32_t stride0)
{
    uint64_t ga = (uint64_t)(uintptr_t)gptr;

    // Group 0 (128b): count=1 (valid), lds_addr, global_addr[56:0], type=2
    u32x4 g0;
    g0.x = 1u;                                        // count=1, is_restore=0
    g0.y = lds_byte_addr;                             // LDS byte address
    g0.z = (uint32_t)(ga & 0xFFFFFFFFu);              // global_addr[31:0]
    g0.w = (uint32_t)((ga >> 32) & 0x01FFFFFFu)       // global_addr[56:32]
         | (2u << 30);                                // type = 2 ("image")

    // Group 1 (256b): mask=0, data_size=4B, dims/tile dims/strides
    i32x8 g1;
    g1[0] = (int)(2u << 16);                                  // data_size=2 (4B)
    g1[1] = (int)((dim0 & 0xFFFFu) << 16);                    // tensor_dim0[15:0] @ [63:48]
    g1[2] = (int)(((dim0 >> 16) & 0xFFFFu)                    // tensor_dim0[31:16]
                  | ((dim1 & 0xFFFFu) << 16));                // tensor_dim1[15:0]
    g1[3] = (int)(((dim1 >> 16) & 0xFFFFu)                    // tensor_dim1[31:16]
                  | ((dim0 & 0xFFFFu) << 16));                // tile_dim0 = full row
    g1[4] = (int)(dim1 & 0xFFFFu);                            // tile_dim1 = dim1, tile_dim2=0
    g1[5] = (int)stride0;                                     // tensor_dim0_stride[31:0]
    g1[6] = 0;                                                // stride hi / dim1_stride lo
    g1[7] = 0;

    // Groups 2/3: unused for 2D tensors (dims 2..4 zero)
    i32x4 g2 = {0, 0, 0, 0};
    i32x4 g3 = {0, 0, 0, 0};

#if defined(__clang_major__) && (__clang_major__ >= 23)
    i32x8 g4 = {0, 0, 0, 0, 0, 0, 0, 0};
    __builtin_amdgcn_tensor_load_to_lds(g0, g1, g2, g3, g4, 0);
#else
    __builtin_amdgcn_tensor_load_to_lds(g0, g1, g2, g3, 0);
#endif
}

// ---------------------------------------------------------------------------
// One workgroup per (b,s) site. TDM stages x (8 KB) and residual (32 KB) into
// LDS asynchronously; 8 waves then compute out[n,d] = post[n]*x[d]
// + sum_m comb[m,n]*residual[m,d] with float4 math and NT b128 stores.
// 40 KB LDS/WG -> up to 8 WGs resident per 320 KB WGP, so one WG's DMA
// overlaps other WGs' compute.
// ---------------------------------------------------------------------------
__global__ __launch_bounds__(256) void
hcpost_fused_kernel(const float* __restrict__ x,
                    const float* __restrict__ residual,
                    const float* __restrict__ post,
                    const float* __restrict__ comb,
                    float* __restrict__ out)
{
    __shared__ __align__(16) float lds_x[Dc];           //  8 KB
    __shared__ __align__(16) float lds_r[Mc * Dc];      // 32 KB

    const uint32_t site = blockIdx.x;                   // b*S + s, 0..16383
    const float* xg = x        + (size_t)site * Dc;
    const float* rg = residual + (size_t)site * (Mc * Dc);
    const float* pg = post     + (size_t)site * Nc;
    const float* cg = comb     + (size_t)site * (Mc * Nc);
    float*       og = out      + (size_t)site * (Nc * Dc);

    // Wave 0 drives the TDM (one issue per wave; EXEC is ignored by tensor
    // ops, so gate at wave granularity). TENSORcnt is per-wave: the issuing
    // wave waits, then the workgroup barrier publishes LDS to all waves.
    if (threadIdx.x < 32u) {
        tdm_load_2d_f32((uint32_t)(uintptr_t)&lds_x[0], xg, Dc, 1u, Dc);
        tdm_load_2d_f32((uint32_t)(uintptr_t)&lds_r[0], rg, Dc, (uint32_t)Mc, Dc);
        __builtin_amdgcn_s_wait_tensorcnt(0);
    }
    __syncthreads();

    // Per-site coefficients: uniform across the block -> scalar (SMEM) loads.
    float p[Nc];
    float c[Mc][Nc];
#pragma unroll
    for (int n = 0; n < Nc; ++n) p[n] = pg[n];
#pragma unroll
    for (int m = 0; m < Mc; ++m)
#pragma unroll
        for (int n = 0; n < Nc; ++n) c[m][n] = cg[m * Nc + n];

    const f32x4* lx = (const f32x4*)lds_x;
    const f32x4* lr = (const f32x4*)lds_r;
    f32x4*       o4 = (f32x4*)og;

    const int NC = Dc / 4;  // 512 float4 chunks per row; 2 per thread
#pragma unroll
    for (int it = 0; it < NC / 256; ++it) {
        const int ci = (int)threadIdx.x + it * 256;

        f32x4 xv = lx[ci];
        f32x4 r0 = lr[0 * NC + ci];
        f32x4 r1 = lr[1 * NC + ci];
        f32x4 r2 = lr[2 * NC + ci];
        f32x4 r3 = lr[3 * NC + ci];

#pragma unroll
        for (int n = 0; n < Nc; ++n) {
            f32x4 acc = p[n] * xv;
            acc += c[0][n] * r0;
            acc += c[1][n] * r1;
            acc += c[2][n] * r2;
            acc += c[3][n] * r3;
            // Output stream (512 MiB) is written once and never re-read:
            // non-temporal to keep it out of WGP$/L2.
            __builtin_nontemporal_store(acc, &o4[n * NC + ci]);
        }
    }
}

extern "C" void kernel_launch(void* const* d_in, const int* in_sizes, int n_in,
                              void* d_out, int out_size, void* d_ws, size_t ws_size,
                              hipStream_t stream) {
    (void)in_sizes; (void)n_in; (void)out_size; (void)d_ws; (void)ws_size;
    const float* x        = (const float*)d_in[0];
    const float* residual = (const float*)d_in[1];
    const float* post     = (const float*)d_in[2];
    const float* comb     = (const float*)d_in[3];
    float*       out      = (float*)d_out;

    dim3 grid(Bc * Sc);   // one workgroup per (b,s) site
    dim3 block(256);
    hcpost_fused_kernel<<<grid, block, 0, stream>>>(x, residual, post, comb, out);
}